// HE_24129126269531
// MI455X (gfx1250) — compile-verified
//
#include <hip/hip_runtime.h>

// CDNA5 / gfx1250: wave32, WMMA f32 16x16x4.
typedef __attribute__((ext_vector_type(2))) float v2f;
typedef __attribute__((ext_vector_type(8))) float v8f;

// One block = 16 output rows. 16 waves: phase 1 -> wave w does top-16/softmax/
// gather for row w; phase 2 -> waves 0..3 compute D[16x64] = A[16x256]xB[256x64]
// with v_wmma_f32_16x16x4_f32 (4 independent accumulator chains for latency
// hiding), where B rows 16i..16i+15 are the gathered parent rows of output
// row i and A[i,16i+j] = softmax weight j of row i.
template <int P>
__global__ __launch_bounds__(512) void topk16_wmma_kernel(
    const float* __restrict__ conn,    // [*, P] row-major
    const int*   __restrict__ ids,     // optional: conn row = ids[r] (else r)
    const float* __restrict__ parent,  // [P, 64]
    float*       __restrict__ out)     // [n, 64]
{
    constexpr int VPL = P / 32;   // values per lane
    constexpr int BS  = 72;       // B-staging row stride (2*BS % 64 == 16 -> no bank clash)
    static_assert(P % 64 == 0, "P must be a multiple of 64");
    __shared__ float lds_w[16 * 16];
    __shared__ float lds_b[256 * BS];

    const int lane    = threadIdx.x & 31;
    const int wave    = threadIdx.x >> 5;
    const int rowbase = blockIdx.x * 16;

    // ---------------- phase 1: top-16 + softmax + gather ----------------
    {
        const int r    = rowbase + wave;
        const int crow = ids ? ids[r] : r;
        const float* rp = conn + (size_t)crow * P;

        // Wide coalesced loads of the connection row (the HBM-roofline term).
        float v[VPL];
        if constexpr (P % 128 == 0) {
            const float4* rp4 = (const float4*)rp;      // global_load_b128
            #pragma unroll
            for (int j = 0; j < VPL / 4; ++j) {
                float4 q = rp4[j * 32 + lane];
                v[4*j+0] = q.x; v[4*j+1] = q.y; v[4*j+2] = q.z; v[4*j+3] = q.w;
            }
        } else {                                        // P == 64
            const float2* rp2 = (const float2*)rp;      // global_load_b64
            float2 q = rp2[lane];
            v[0] = q.x; v[1] = q.y;
        }

        unsigned taken = 0u;
        float m0 = 0.f, s = 0.f;
        #pragma unroll 1
        for (int t = 0; t < 16; ++t) {
            // lane-local max over not-yet-taken values
            float lm = -3.4e38f; int lj = 0;
            #pragma unroll
            for (int j = 0; j < VPL; ++j) {
                bool ok = !((taken >> j) & 1u) && (v[j] > lm);
                lm = ok ? v[j] : lm;
                lj = ok ? j : lj;
            }
            // wave-wide argmax (wave32)
            float m = lm; int src = lane;
            #pragma unroll
            for (int off = 16; off > 0; off >>= 1) {
                float om = __shfl_xor(m, off, 32);
                int   os = __shfl_xor(src, off, 32);
                bool  tk = (om > m) || (om == m && os < src);
                m   = tk ? om : m;
                src = tk ? os : src;
            }
            int ljw = __shfl(lj, src, 32);
            if (lane == src) taken |= (1u << lj);

            if (t == 0) m0 = m;              // first pick == row max
            float e = __expf(m - m0);        // streaming softmax numerator
            s += e;
            if (lane == 0) lds_w[wave * 16 + t] = e;

            // recover the global element index of the winner
            int pidx;
            if constexpr (P % 128 == 0) {
                pidx = ((ljw >> 2) * 128) + (src << 2) + (ljw & 3);
            } else {
                pidx = (src << 1) + ljw;
            }

            // gather the selected parent row into B-staging (row wave*16+t)
            const float2* pr2 = (const float2*)(parent + (size_t)pidx * 64);
            float2* dst2 = (float2*)&lds_b[(wave * 16 + t) * BS];
            dst2[lane] = pr2[lane];          // b64 load + ds_store_b64
        }
        float inv = 1.f / s;
        if (lane < 16) lds_w[wave * 16 + lane] *= inv;  // normalize weights
    }
    __syncthreads();

    // ---------------- phase 2: WMMA f32 16x16x4, 4 accumulator chains ----------------
    if (wave < 4) {
        const int ntile = wave;          // 16-column tile of the 64-wide output
        const int N     = lane & 15;     // also M index for A fragment
        const int khalf = lane >> 4;     // 0: K={0,1}, 1: K={2,3} within chunk

        auto step = [&](int kc, v8f c) -> v8f {
            const int owner = kc >> 2;             // output row owning this K-chunk
            const int jbase = ((kc & 3) << 2) + (khalf << 1);
            const float w0 = lds_w[owner * 16 + jbase];
            const float w1 = lds_w[owner * 16 + jbase + 1];
            v2f a, b;
            a.x = (N == owner) ? w0 : 0.f;         // A[M, 4kc+2*khalf]
            a.y = (N == owner) ? w1 : 0.f;         // A[M, 4kc+2*khalf+1]
            const int kr = 4 * kc + 2 * khalf;     // B rows for this lane half
            b.x = lds_b[(kr    ) * BS + ntile * 16 + N];
            b.y = lds_b[(kr + 1) * BS + ntile * 16 + N];
            // (neg_a, A, neg_b, B, c_mod, C, reuse_a, reuse_b)
            return __builtin_amdgcn_wmma_f32_16x16x4_f32(false, a, false, b,
                                                         (short)0, c, false, false);
        };

        v8f c0 = {}, c1 = {}, c2 = {}, c3 = {};
        #pragma unroll 1
        for (int kc = 0; kc < 64; kc += 4) {       // K = 256 in steps of 4x4
            c0 = step(kc + 0, c0);                 // independent chains: WMMA
            c1 = step(kc + 1, c1);                 // latency hidden by ILP
            c2 = step(kc + 2, c2);
            c3 = step(kc + 3, c3);
        }
        #pragma unroll
        for (int v8 = 0; v8 < 8; ++v8) {
            const int mrow = v8 + 8 * khalf;       // C/D layout: VGPR v -> M=v / v+8
            const float d = (c0[v8] + c1[v8]) + (c2[v8] + c3[v8]);
            out[(size_t)(rowbase + mrow) * 64 + ntile * 16 + N] = d;
        }
    }
}

extern "C" void kernel_launch(void* const* d_in, const int* in_sizes, int n_in,
                              void* d_out, int out_size, void* d_ws, size_t ws_size,
                              hipStream_t stream) {
    (void)n_in; (void)out_size; (void)ws_size;
    const int*   ids  = (const int*)  d_in[0];   // [8192] int32
    const float* c1   = (const float*)d_in[1];   // [200000, 1024]
    const float* c2   = (const float*)d_in[2];   // [1024, 256]
    const float* c3   = (const float*)d_in[3];   // [256, 64]
    const float* root = (const float*)d_in[4];   // [64, 64]
    float* out = (float*)d_out;                  // [8192, 64]

    float* e2 = (float*)d_ws;        // [256, 64]  = 64 KB
    float* e1 = e2 + 256 * 64;       // [1024, 64] = 256 KB

    const int batch = in_sizes[0];   // 8192 (multiple of 16)

    topk16_wmma_kernel<64>  <<<256  / 16, 512, 0, stream>>>(c3, nullptr, root, e2);
    topk16_wmma_kernel<256> <<<1024 / 16, 512, 0, stream>>>(c2, nullptr, e2,   e1);
    topk16_wmma_kernel<1024><<<batch / 16, 512, 0, stream>>>(c1, ids,     e1,  out);
}